// GAT_model1_32779190403779
// MI455X (gfx1250) — compile-verified
//
#include <hip/hip_runtime.h>
#include <math.h>

// ---------------------------------------------------------------------------
// GAT (2 layers, 4 heads then 1 head) + global mean pool + linear classifier.
// MI455X / gfx1250: wave32, WMMA f32 16x16x4 with LDS-staged B panels.
// ---------------------------------------------------------------------------

typedef __attribute__((ext_vector_type(2))) float v2f;
typedef __attribute__((ext_vector_type(8))) float v8f;

#define NEG_SLOPE 0.2f
#define KC 32            // K-chunk staged in LDS
#define LDSB 80          // padded LDS row stride (floats): halves hit disjoint banks

__device__ __forceinline__ float atomicMaxF(float* addr, float val) {
  // Works for mixed signs given init to -inf.
  if (val >= 0.0f)
    return __int_as_float(atomicMax((int*)addr, __float_as_int(val)));
  else
    return __uint_as_float(atomicMin((unsigned int*)addr, __float_as_uint(val)));
}

// ---------------------------------------------------------------------------
// C[M,Nc] = A[M,K] @ B[K,Nc], fp32, via V_WMMA_F32_16X16X4_F32.
// Block = 256 threads (8 waves) -> 128x64 output tile; B panel shared via LDS.
// Requirements: M%16==0, K%32==0, Nc%64==0. Grid: (Nc/64, ceil(M/128)).
// A-frag (16x4 f32): lanes 0-15 -> M=lane, K={k,k+1}; lanes 16-31 -> K={k+2,k+3}.
// B-frag (4x16 f32): lanes 0-15 -> N=lane, K={k,k+1}; lanes 16-31 -> K={k+2,k+3}.
// C/D: VGPR v: lanes 0-15 -> M=v, lanes 16-31 -> M=v+8; N=lane%16.
// ---------------------------------------------------------------------------
__global__ __launch_bounds__(256)
void gemm_wmma_f32(const float* __restrict__ A, const float* __restrict__ B,
                   float* __restrict__ C, int M, int K, int Nc) {
  __shared__ float bs[KC * LDSB];

  const int wave = threadIdx.x >> 5;
  const int lane = threadIdx.x & 31;
  const int M16  = M >> 4;
  const int mt   = blockIdx.y * 8 + wave;      // 16-row tile index for this wave
  const bool valid = (mt < M16);
  const int m0   = mt << 4;
  const int n0   = blockIdx.x << 6;
  const int half = lane >> 4;                  // 0: K pair {0,1}, 1: K pair {2,3}
  const int r    = lane & 15;

  v8f acc0 = {}, acc1 = {}, acc2 = {}, acc3 = {};
  const float* arow = A + (size_t)((valid ? m0 : 0) + r) * K + (half << 1);

  for (int kc = 0; kc < K; kc += KC) {
    // Cooperative, coalesced load of B[kc..kc+KC)[n0..n0+64) into LDS.
#pragma unroll
    for (int i = 0; i < (KC * 64) / 256; ++i) {
      int idx = threadIdx.x + 256 * i;
      int row = idx >> 6, col = idx & 63;
      bs[row * LDSB + col] = B[(size_t)(kc + row) * Nc + n0 + col];
    }
    __syncthreads();

    if (valid) {
#pragma unroll
      for (int kk = 0; kk < KC; kk += 4) {
        v2f a;
        a.x = arow[kc + kk];
        a.y = arow[kc + kk + 1];
        const float* b0 = &bs[(kk + (half << 1)) * LDSB + r];
        const float* b1 = b0 + LDSB;
        v2f b;
        b.x = b0[0];  b.y = b1[0];
        acc0 = __builtin_amdgcn_wmma_f32_16x16x4_f32(false, a, false, b, (short)0, acc0, false, false);
        b.x = b0[16]; b.y = b1[16];
        acc1 = __builtin_amdgcn_wmma_f32_16x16x4_f32(false, a, false, b, (short)0, acc1, false, false);
        b.x = b0[32]; b.y = b1[32];
        acc2 = __builtin_amdgcn_wmma_f32_16x16x4_f32(false, a, false, b, (short)0, acc2, false, false);
        b.x = b0[48]; b.y = b1[48];
        acc3 = __builtin_amdgcn_wmma_f32_16x16x4_f32(false, a, false, b, (short)0, acc3, false, false);
      }
    }
    __syncthreads();
  }

  if (valid) {
    float* crow = C + (size_t)(m0 + half * 8) * Nc + n0 + r;
#pragma unroll
    for (int v = 0; v < 8; ++v) {
      crow[(size_t)v * Nc + 0]  = acc0[v];
      crow[(size_t)v * Nc + 16] = acc1[v];
      crow[(size_t)v * Nc + 32] = acc2[v];
      crow[(size_t)v * Nc + 48] = acc3[v];
    }
  }
}

// ---------------------------------------------------------------------------
// Per-(node,head) attention logits: al_s[n,h] = dot(h[n,h,:], a_src[h,:]) etc.
// One wave per (n,h); C=128 -> 4 elems/lane + shuffle reduction.
// ---------------------------------------------------------------------------
__global__ void att_logits(const float* __restrict__ h, const float* __restrict__ a_src,
                           const float* __restrict__ a_dst, float* __restrict__ al_s,
                           float* __restrict__ al_d, int N, int H, int C) {
  int wid  = (blockIdx.x * blockDim.x + threadIdx.x) >> 5;
  int lane = threadIdx.x & 31;
  if (wid >= N * H) return;
  int n = wid / H, hh = wid % H;
  const float* hp = h + (size_t)n * H * C + (size_t)hh * C;
  float ss = 0.f, sd = 0.f;
  for (int c = lane; c < C; c += 32) {
    float v = hp[c];
    ss += v * a_src[hh * C + c];
    sd += v * a_dst[hh * C + c];
  }
#pragma unroll
  for (int off = 16; off > 0; off >>= 1) {
    ss += __shfl_down(ss, off, 32);
    sd += __shfl_down(sd, off, 32);
  }
  if (lane == 0) { al_s[wid] = ss; al_d[wid] = sd; }
}

// ---------------------------------------------------------------------------
// Edge helpers (edges 0..E-1 from edge_index, E..E+N-1 are self-loops).
// ---------------------------------------------------------------------------
__device__ __forceinline__ void edge_sd(const int* __restrict__ ei, int e, int E,
                                        int& s, int& d) {
  if (e < E) { s = ei[e]; d = ei[E + e]; }
  else       { s = e - E; d = e - E; }
}

__global__ void edge_max(const int* __restrict__ ei, int E, int N, int H,
                         const float* __restrict__ als, const float* __restrict__ ald,
                         float* __restrict__ m) {
  long t = (long)blockIdx.x * blockDim.x + threadIdx.x;
  long total = (long)(E + N) * H;
  if (t >= total) return;
  int e = (int)(t / H), hh = (int)(t % H);
  int s, d; edge_sd(ei, e, E, s, d);
  float v = als[s * H + hh] + ald[d * H + hh];
  v = (v > 0.f) ? v : NEG_SLOPE * v;
  atomicMaxF(&m[d * H + hh], v);
}

__global__ void edge_denom(const int* __restrict__ ei, int E, int N, int H,
                           const float* __restrict__ als, const float* __restrict__ ald,
                           const float* __restrict__ m, float* __restrict__ dn) {
  long t = (long)blockIdx.x * blockDim.x + threadIdx.x;
  long total = (long)(E + N) * H;
  if (t >= total) return;
  int e = (int)(t / H), hh = (int)(t % H);
  int s, d; edge_sd(ei, e, E, s, d);
  float v = als[s * H + hh] + ald[d * H + hh];
  v = (v > 0.f) ? v : NEG_SLOPE * v;
  atomicAdd(&dn[d * H + hh], expf(v - m[d * H + hh]));
}

// One wave per (edge, head): scatter h[src,h,:]*alpha into out[dst,h,:].
__global__ void edge_agg(const int* __restrict__ ei, int E, int N, int H, int C,
                         const float* __restrict__ als, const float* __restrict__ ald,
                         const float* __restrict__ m, const float* __restrict__ dn,
                         const float* __restrict__ h, float* __restrict__ out) {
  long wid = (long)(blockIdx.x * blockDim.x + threadIdx.x) >> 5;
  int lane = threadIdx.x & 31;
  long total = (long)(E + N) * H;
  if (wid >= total) return;
  int e = (int)(wid / H), hh = (int)(wid % H);
  int s, d; edge_sd(ei, e, E, s, d);
  float v = als[s * H + hh] + ald[d * H + hh];
  v = (v > 0.f) ? v : NEG_SLOPE * v;
  float alpha = expf(v - m[d * H + hh]) / (dn[d * H + hh] + 1e-16f);
  const float* hs = h + (size_t)s * H * C + (size_t)hh * C;
  float*       od = out + (size_t)d * H * C + (size_t)hh * C;
  for (int c = lane; c < C; c += 32) atomicAdd(&od[c], hs[c] * alpha);
}

// ---------------------------------------------------------------------------
// Elementwise / init / pooling / classifier tails.
// ---------------------------------------------------------------------------
__global__ void fill_f32(float* __restrict__ p, float v, long n) {
  long i = (long)blockIdx.x * blockDim.x + threadIdx.x;
  if (i < n) p[i] = v;
}

__global__ void init_bias(float* __restrict__ out, const float* __restrict__ b,
                          int N, int D) {
  long i = (long)blockIdx.x * blockDim.x + threadIdx.x;
  if (i < (long)N * D) out[i] = b[i % D];
}

__global__ void relu_ip(float* __restrict__ p, long n) {
  long i = (long)blockIdx.x * blockDim.x + threadIdx.x;
  if (i < n) p[i] = fmaxf(p[i], 0.f);
}

__global__ void pool_count(const int* __restrict__ batch, float* __restrict__ cnt, int N) {
  int i = blockIdx.x * blockDim.x + threadIdx.x;
  if (i < N) atomicAdd(&cnt[batch[i]], 1.0f);
}

__global__ void pool_sum(const int* __restrict__ batch, const float* __restrict__ h,
                         float* __restrict__ sums, int N, int C) {
  long t = (long)blockIdx.x * blockDim.x + threadIdx.x;
  if (t >= (long)N * C) return;
  int n = (int)(t / C), c = (int)(t % C);
  atomicAdd(&sums[batch[n] * C + c], h[t]);
}

__global__ void classifier(const float* __restrict__ sums, const float* __restrict__ cnt,
                           const float* __restrict__ Wc, const float* __restrict__ bc,
                           float* __restrict__ out, int G, int C, int O) {
  int t = blockIdx.x * blockDim.x + threadIdx.x;
  if (t >= G * O) return;
  int g = t / O, o = t % O;
  float inv = 1.0f / fmaxf(cnt[g], 1.0f);
  float s = bc[o];
  for (int c = 0; c < C; ++c) s += (sums[g * C + c] * inv) * Wc[c * O + o];
  out[t] = s;
}

// ---------------------------------------------------------------------------
// Host launch
// ---------------------------------------------------------------------------
static inline int cdiv(long a, long b) { return (int)((a + b - 1) / b); }

extern "C" void kernel_launch(void* const* d_in, const int* in_sizes, int n_in,
                              void* d_out, int out_size, void* d_ws, size_t ws_size,
                              hipStream_t stream) {
  (void)n_in; (void)out_size; (void)ws_size;

  const int F  = 768;                     // input features
  const int H1 = 4, C = 128, D1 = H1 * C; // layer-1 heads, channels, 512
  const int G  = 16, O = 10;              // graphs, classes
  const int N  = in_sizes[0] / F;         // 10000
  const int E  = in_sizes[1] / 2;         // 160000
  const int EE = E + N;                   // + self loops

  const float* x    = (const float*)d_in[0];
  const int*   ei   = (const int*)  d_in[1];
  const int*   bat  = (const int*)  d_in[2];
  const float* W1   = (const float*)d_in[3];
  const float* as1  = (const float*)d_in[4];
  const float* ad1  = (const float*)d_in[5];
  const float* b1   = (const float*)d_in[6];
  const float* W2   = (const float*)d_in[7];
  const float* as2  = (const float*)d_in[8];
  const float* ad2  = (const float*)d_in[9];
  const float* b2   = (const float*)d_in[10];
  const float* Wc   = (const float*)d_in[11];
  const float* bc   = (const float*)d_in[12];
  float*       out  = (float*)d_out;

  // Workspace carve-up (floats).
  float* W = (float*)d_ws;
  size_t off = 0;
  float* h1   = W + off; off += (size_t)N * D1;
  float* o1   = W + off; off += (size_t)N * D1;
  float* h2   = W + off; off += (size_t)N * C;
  float* o2   = W + off; off += (size_t)N * C;
  float* als1 = W + off; off += (size_t)N * H1;
  float* ald1 = W + off; off += (size_t)N * H1;
  float* m1   = W + off; off += (size_t)N * H1;
  float* dn1  = W + off; off += (size_t)N * H1;
  float* als2 = W + off; off += (size_t)N;
  float* ald2 = W + off; off += (size_t)N;
  float* m2   = W + off; off += (size_t)N;
  float* dn2  = W + off; off += (size_t)N;
  float* psum = W + off; off += (size_t)G * C;
  float* pcnt = W + off; off += (size_t)G;

  const int TB = 256;
  const int M16 = N / 16;                 // 625 (N % 16 == 0)

  // ----- Layer 1 -----
  // h1 = x @ W1   (M=N, K=768, Nc=512)
  {
    dim3 grid(D1 / 64, cdiv(M16, 8));
    gemm_wmma_f32<<<grid, TB, 0, stream>>>(x, W1, h1, N, F, D1);
  }
  att_logits<<<cdiv((long)N * H1 * 32, TB), TB, 0, stream>>>(h1, as1, ad1, als1, ald1, N, H1, C);
  fill_f32<<<cdiv((long)N * H1, TB), TB, 0, stream>>>(m1, -INFINITY, (long)N * H1);
  fill_f32<<<cdiv((long)N * H1, TB), TB, 0, stream>>>(dn1, 0.0f, (long)N * H1);
  init_bias<<<cdiv((long)N * D1, TB), TB, 0, stream>>>(o1, b1, N, D1);  // out = segsum + b
  edge_max  <<<cdiv((long)EE * H1, TB), TB, 0, stream>>>(ei, E, N, H1, als1, ald1, m1);
  edge_denom<<<cdiv((long)EE * H1, TB), TB, 0, stream>>>(ei, E, N, H1, als1, ald1, m1, dn1);
  edge_agg  <<<cdiv((long)EE * H1 * 32, TB), TB, 0, stream>>>(ei, E, N, H1, C, als1, ald1, m1, dn1, h1, o1);
  relu_ip<<<cdiv((long)N * D1, TB), TB, 0, stream>>>(o1, (long)N * D1);

  // ----- Layer 2 (1 head) -----
  // h2 = relu(o1) @ W2   (M=N, K=512, Nc=128)
  {
    dim3 grid(C / 64, cdiv(M16, 8));
    gemm_wmma_f32<<<grid, TB, 0, stream>>>(o1, W2, h2, N, D1, C);
  }
  att_logits<<<cdiv((long)N * 32, TB), TB, 0, stream>>>(h2, as2, ad2, als2, ald2, N, 1, C);
  fill_f32<<<cdiv((long)N, TB), TB, 0, stream>>>(m2, -INFINITY, (long)N);
  fill_f32<<<cdiv((long)N, TB), TB, 0, stream>>>(dn2, 0.0f, (long)N);
  init_bias<<<cdiv((long)N * C, TB), TB, 0, stream>>>(o2, b2, N, C);
  edge_max  <<<cdiv((long)EE, TB), TB, 0, stream>>>(ei, E, N, 1, als2, ald2, m2);
  edge_denom<<<cdiv((long)EE, TB), TB, 0, stream>>>(ei, E, N, 1, als2, ald2, m2, dn2);
  edge_agg  <<<cdiv((long)EE * 32, TB), TB, 0, stream>>>(ei, E, N, 1, C, als2, ald2, m2, dn2, h2, o2);
  relu_ip<<<cdiv((long)N * C, TB), TB, 0, stream>>>(o2, (long)N * C);

  // ----- Global mean pool + classifier -----
  fill_f32<<<cdiv((long)G * C, TB), TB, 0, stream>>>(psum, 0.0f, (long)G * C);
  fill_f32<<<1, TB, 0, stream>>>(pcnt, 0.0f, (long)G);
  pool_count<<<cdiv((long)N, TB), TB, 0, stream>>>(bat, pcnt, N);
  pool_sum<<<cdiv((long)N * C, TB), TB, 0, stream>>>(bat, o2, psum, N, C);
  classifier<<<1, TB, 0, stream>>>(psum, pcnt, Wc, bc, out, G, C, O);
}